// VQ_25881472925808
// MI455X (gfx1250) — compile-verified
//
#include <hip/hip_runtime.h>
#include <stdint.h>

typedef __attribute__((ext_vector_type(2))) float v2f;
typedef __attribute__((ext_vector_type(8))) float v8f;

#define N_FEAT 32768
#define D_DIM  256
#define K_CODE 4096

// ---- workspace layout (bytes) ----
#define WS_FWD  0                          // N x u64 packed (distKey<<32 | k)
#define WS_REV  (N_FEAT * 8)               // K x u64 packed (distKey<<32 | n)
#define WS_F2   (WS_REV + K_CODE * 8)      // N x f32
#define WS_C2   (WS_F2 + N_FEAT * 4)       // K x f32
#define WS_CNT  (WS_C2 + K_CODE * 4)       // K x u32 assignment counts
#define WS_SCAL (WS_CNT + K_CODE * 4)      // scalars: [0]=cb_loss f32, [1]=un_num f32,
                                           //          [2]=un_cnt u32, [3]=assigned u32, [4]=mask_sum u32

__device__ __forceinline__ unsigned orderKey(float f) {
    unsigned b = __float_as_uint(f);
    return (b & 0x80000000u) ? ~b : (b | 0x80000000u);   // monotonic float -> uint
}
__device__ __forceinline__ unsigned long long min64(unsigned long long a, unsigned long long b) {
    return a < b ? a : b;
}
__device__ __forceinline__ unsigned long long shflxor_u64(unsigned long long v, int m) {
    int lo = __shfl_xor((int)(unsigned)(v & 0xffffffffull), m, 32);
    int hi = __shfl_xor((int)(unsigned)(v >> 32), m, 32);
    return (((unsigned long long)(unsigned)hi) << 32) | (unsigned)lo;
}

// ------------------------------------------------------------------ init
__global__ __launch_bounds__(256) void vq_init(const float* __restrict__ F,
                                               const float* __restrict__ P,
                                               void* __restrict__ ws) {
    int tid = blockIdx.x * blockDim.x + threadIdx.x;
    unsigned long long* fwd = (unsigned long long*)((char*)ws + WS_FWD);
    unsigned long long* rev = (unsigned long long*)((char*)ws + WS_REV);
    float* f2 = (float*)((char*)ws + WS_F2);
    float* c2 = (float*)((char*)ws + WS_C2);
    unsigned* cnt = (unsigned*)((char*)ws + WS_CNT);
    float* scalf = (float*)((char*)ws + WS_SCAL);
    unsigned* scalu = (unsigned*)((char*)ws + WS_SCAL);

    if (tid < K_CODE) {
        const float* p = P + (size_t)tid * D_DIM;
        float s = 0.f;
        for (int d = 0; d < D_DIM; ++d) { float c = 10.f * p[d]; s += c * c; }
        c2[tid] = s;
        rev[tid] = ~0ull;
        cnt[tid] = 0u;
    }
    if (tid < N_FEAT) {
        const float* f = F + (size_t)tid * D_DIM;
        float s = 0.f;
        for (int d = 0; d < D_DIM; ++d) { float x = f[d]; s += x * x; }
        f2[tid] = s;
        fwd[tid] = ~0ull;
    }
    if (tid == 0) { scalf[0] = 0.f; scalf[1] = 0.f; scalu[2] = 0u; scalu[3] = 0u; scalu[4] = 0u; }
}

// ------------------------------------------------------------------ fused GEMM + dual argmin
// grid = (K/256, N/128), block = 256 (8 waves). Wave w: 16 rows; loops 16 tiles of 16 codes.
__global__ __launch_bounds__(256) void vq_gemm(const float* __restrict__ F,
                                               const unsigned char* __restrict__ mask,
                                               const float* __restrict__ P,
                                               void* __restrict__ ws) {
    const int lane = threadIdx.x & 31;
    const int wave = threadIdx.x >> 5;
    const int row0 = blockIdx.y * 128 + wave * 16;
    const int col0 = blockIdx.x * 256;
    const int m = lane & 15;
    const int khalf = (lane >> 4) << 1;    // K offset 0 or 2 within the 16x4 step
    const int halfoff = (lane >> 4) << 3;  // D-matrix row offset 0 or 8

    const float* f2 = (const float*)((const char*)ws + WS_F2);
    const float* c2 = (const float*)((const char*)ws + WS_C2);
    unsigned long long* fwd = (unsigned long long*)((char*)ws + WS_FWD);
    unsigned long long* rev = (unsigned long long*)((char*)ws + WS_REV);

    // Preload A panel: 16 rows x 256 d -> 64 float2 per lane, pre-scaled by 10
    // ((10f)·p == f·(10p) up to ulp; moves the codebook scale out of the hot loop)
    v2f areg[64];
    const float* frow = F + (size_t)(row0 + m) * D_DIM;
#pragma unroll
    for (int dt = 0; dt < 64; ++dt) {
        v2f a = *(const v2f*)(frow + dt * 4 + khalf);
        areg[dt] = a * 10.f;
    }

    float f2r[8];
    unsigned mk[8];
#pragma unroll
    for (int v = 0; v < 8; ++v) {
        int rn = row0 + v + halfoff;
        f2r[v] = f2[rn];
        mk[v] = mask[rn];
    }

    unsigned long long rowkey[8];
#pragma unroll
    for (int v = 0; v < 8; ++v) rowkey[v] = ~0ull;

    for (int kt = 0; kt < 16; ++kt) {
        const int kbase = col0 + kt * 16;
        const float* brow = P + (size_t)(kbase + m) * D_DIM;  // B col = lane&15 -> code row
        v8f acc0 = {};
        v8f acc1 = {};  // two chains so back-to-back WMMAs pipeline within one wave
#pragma unroll
        for (int dt = 0; dt < 64; ++dt) {
            v2f b = *(const v2f*)(brow + dt * 4 + khalf);
            if (dt & 1)
                acc1 = __builtin_amdgcn_wmma_f32_16x16x4_f32(
                    false, areg[dt], false, b, (short)0, acc1, false, false);
            else
                acc0 = __builtin_amdgcn_wmma_f32_16x16x4_f32(
                    false, areg[dt], false, b, (short)0, acc0, false, false);
        }
        const int colk = kbase + m;
        const float c2k = c2[colk];
        unsigned long long colkey = ~0ull;
#pragma unroll
        for (int v = 0; v < 8; ++v) {
            float fc = acc0[v] + acc1[v];
            float dist = mk[v] ? (f2r[v] + c2k - 2.0f * fc) : __builtin_huge_valf();
            unsigned long long base = ((unsigned long long)orderKey(dist)) << 32;
            rowkey[v] = min64(rowkey[v], base | (unsigned)colk);
            colkey = min64(colkey, base | (unsigned)(row0 + v + halfoff));
        }
        // column (per-code) min: combine lane halves, lanes 0..15 hold 16 distinct codes
        colkey = min64(colkey, shflxor_u64(colkey, 16));
        if (lane < 16) atomicMin(&rev[colk], colkey);
    }

    // row (per-feature) min: reduce across the 16 lanes of each half, then one atomic per row
#pragma unroll
    for (int v = 0; v < 8; ++v) {
        unsigned long long k = rowkey[v];
        k = min64(k, shflxor_u64(k, 1));
        k = min64(k, shflxor_u64(k, 2));
        k = min64(k, shflxor_u64(k, 4));
        k = min64(k, shflxor_u64(k, 8));
        if (m == 0) atomicMin(&fwd[row0 + v + halfoff], k);
    }
}

// ------------------------------------------------------------------ per-row outputs + losses
// wave per feature row. grid = N/8, block = 256.
__global__ __launch_bounds__(256) void vq_assign(const float* __restrict__ F,
                                                 const unsigned char* __restrict__ mask,
                                                 const float* __restrict__ P,
                                                 void* __restrict__ ws,
                                                 float* __restrict__ out) {
    const int lane = threadIdx.x & 31;
    const int wave = threadIdx.x >> 5;
    const int n = blockIdx.x * 8 + wave;

    const unsigned long long* fwd = (const unsigned long long*)((const char*)ws + WS_FWD);
    unsigned* cnt = (unsigned*)((char*)ws + WS_CNT);
    float* scalf = (float*)((char*)ws + WS_SCAL);
    unsigned* scalu = (unsigned*)((char*)ws + WS_SCAL);

    const unsigned kstar = (unsigned)(fwd[n] & 0xffffffffull);
    const float* crow = P + (size_t)kstar * D_DIM;
    const float* frow = F + (size_t)n * D_DIM;

    float s = 0.f;
#pragma unroll
    for (int j = 0; j < 8; ++j) {
        int d = j * 32 + lane;
        float c = 10.f * crow[d];
        out[(size_t)n * D_DIM + d] = c;  // out_features == codebook[assign_fwd] numerically
        float df = c - frow[d];
        s += df * df;
    }
    for (int off = 1; off < 32; off <<= 1) s += __shfl_xor(s, off, 32);

    if (lane == 0) {
        out[(size_t)N_FEAT * D_DIM + n] = (float)kstar;  // assign_fwd as float
        unsigned mn = mask[n] ? 1u : 0u;
        if (mn) {
            atomicAdd(&scalf[0], s * (1.0f / D_DIM));  // codebook/commitment loss numerator
            atomicAdd(&cnt[kstar], 1u);
        }
        atomicAdd(&scalu[4], mn);  // mask sum
    }
}

// ------------------------------------------------------------------ per-code unassigned loss
// wave per code. grid = K/8, block = 256.
__global__ __launch_bounds__(256) void vq_codes(const float* __restrict__ F,
                                                const float* __restrict__ P,
                                                void* __restrict__ ws) {
    const int lane = threadIdx.x & 31;
    const int wave = threadIdx.x >> 5;
    const int k = blockIdx.x * 8 + wave;

    const unsigned long long* rev = (const unsigned long long*)((const char*)ws + WS_REV);
    const unsigned* cnt = (const unsigned*)((const char*)ws + WS_CNT);
    float* scalf = (float*)((char*)ws + WS_SCAL);
    unsigned* scalu = (unsigned*)((char*)ws + WS_SCAL);

    const unsigned c = cnt[k];
    if (lane == 0 && c > 0) atomicAdd(&scalu[3], 1u);  // assigned count
    if (c < 1) {
        const unsigned nstar = (unsigned)(rev[k] & 0xffffffffull);
        const float* crow = P + (size_t)k * D_DIM;
        const float* frow = F + (size_t)nstar * D_DIM;
        float s = 0.f;
#pragma unroll
        for (int j = 0; j < 8; ++j) {
            int d = j * 32 + lane;
            float df = 10.f * crow[d] - frow[d];
            s += df * df;
        }
        for (int off = 1; off < 32; off <<= 1) s += __shfl_xor(s, off, 32);
        if (lane == 0) {
            atomicAdd(&scalf[1], s * (1.0f / D_DIM));
            atomicAdd(&scalu[2], 1u);
        }
    }
}

// ------------------------------------------------------------------ final scalars
__global__ void vq_final(void* __restrict__ ws, float* __restrict__ out) {
    const float* scalf = (const float*)((const char*)ws + WS_SCAL);
    const unsigned* scalu = (const unsigned*)((const char*)ws + WS_SCAL);
    unsigned msum = scalu[4];
    float denom = (float)(msum > 1u ? msum : 1u);
    float cb = scalf[0] / denom;
    unsigned ucnt = scalu[2];
    float un = scalf[1] / (float)(ucnt > 1u ? ucnt : 1u);
    size_t base = (size_t)N_FEAT * D_DIM + N_FEAT;
    out[base + 0] = cb;                               // codebook_loss
    out[base + 1] = cb;                               // commitment_loss (numerically identical)
    out[base + 2] = un;                               // unassigned_loss
    out[base + 3] = (float)scalu[3] / (float)K_CODE;  // unassigned_percent
}

// ------------------------------------------------------------------ launch
extern "C" void kernel_launch(void* const* d_in, const int* in_sizes, int n_in,
                              void* d_out, int out_size, void* d_ws, size_t ws_size,
                              hipStream_t stream) {
    const float* F = (const float*)d_in[0];                  // features (N, D) f32
    const unsigned char* M = (const unsigned char*)d_in[1];  // mask (N,) bool (1 byte)
    const float* P = (const float*)d_in[2];                  // codebook_param (K, D) f32
    float* out = (float*)d_out;

    vq_init<<<N_FEAT / 256, 256, 0, stream>>>(F, P, d_ws);

    dim3 grid(K_CODE / 256, N_FEAT / 128);
    vq_gemm<<<grid, 256, 0, stream>>>(F, M, P, d_ws);

    vq_assign<<<N_FEAT / 8, 256, 0, stream>>>(F, M, P, d_ws, out);
    vq_codes<<<K_CODE / 8, 256, 0, stream>>>(F, P, d_ws);
    vq_final<<<1, 1, 0, stream>>>(d_ws, out);
}